// BatchedRegionProposalNetwork_26680336843630
// MI455X (gfx1250) — compile-verified
//
#include <hip/hip_runtime.h>
#include <math.h>

// ---------------- problem constants ----------------
#define A_TOTAL   261888
#define BATCH     16
#define PRE_N     1000
#define NPAD      1024           // padded candidate count (32 words of bits)
#define IMG_F     1024.0f
#define IOU_C     0.41176470588f // 0.7/(1+0.7): inter > IOU_C*(aI+aJ) <=> IoU>0.7
#define MIN_SZ    1e-3f
#define DWH_CLAMP 4.135166556742356f   // log(1000/16)
#define LVL_OFF   1025.0f
#define BIG_AREA  1e30f          // sentinel area for invalid/padded boxes

typedef float v2f __attribute__((ext_vector_type(2)));
typedef float v8f __attribute__((ext_vector_type(8)));
typedef int   v4i __attribute__((ext_vector_type(4)));

// ---------------- CDNA5 feature guards ----------------
#if defined(__gfx1250__) && __has_builtin(__builtin_amdgcn_global_load_async_to_lds_b128) && __has_builtin(__builtin_amdgcn_s_wait_asynccnt)
#define USE_ASYNC_LDS 1
#else
#define USE_ASYNC_LDS 0
#endif

#if defined(__gfx1250__) && __has_builtin(__builtin_amdgcn_wmma_f32_16x16x4_f32)
#define USE_WMMA_F32 1
#else
#define USE_WMMA_F32 0
#endif

#if USE_ASYNC_LDS
__device__ __forceinline__ void ldsCopyB128(void* lds_dst, const void* gsrc) {
  // gfx1250 GLOBAL_LOAD_ASYNC_TO_LDS_B128: per-lane 16B copy, tracked by ASYNCcnt.
  // Builtin params are width-typed: (int4 addrspace(1)*, int4 addrspace(3)*, imm, imm)
  __builtin_amdgcn_global_load_async_to_lds_b128(
      (__attribute__((address_space(1))) v4i*)(void*)gsrc,
      (__attribute__((address_space(3))) v4i*)lds_dst, 0, 0);
}
__device__ __forceinline__ void ldsCopyWait() { __builtin_amdgcn_s_wait_asynccnt(0); }
#endif

// monotonic float->uint key (larger float => larger key)
__device__ __forceinline__ unsigned fkey(float f) {
  unsigned u = __float_as_uint(f);
  return (u & 0x80000000u) ? ~u : (u | 0x80000000u);
}

// =====================================================================
// Kernel 1: exact top-1000 (sorted desc) per batch via 3-round radix
// select (11+11+10 bits) + 1024-element bitonic sort in LDS.
// One 1024-thread block per batch.
// =====================================================================
__global__ __launch_bounds__(1024) void topk_kernel(const float* __restrict__ obj,
                                                    float* __restrict__ tscore,
                                                    unsigned* __restrict__ tidx) {
  __shared__ unsigned hist[2048];
  __shared__ float    s_sc[NPAD];
  __shared__ unsigned s_idx[NPAD];
  __shared__ unsigned s_sel, s_above, s_cnt_gt, s_cnt_eq;

  const int tid  = threadIdx.x;
  const int b    = blockIdx.x;
  const size_t base = (size_t)b * A_TOTAL;

  const int lo_r[3]   = {21, 10, 0};
  const int bits_r[3] = {11, 11, 10};
  const int nb_r[3]   = {2048, 2048, 1024};

  unsigned prefix = 0, need = PRE_N;
  int prefShift = 32;  // 32 => no prefix filter on round 0

  for (int r = 0; r < 3; ++r) {
    const int lo = lo_r[r];
    const unsigned nb = nb_r[r];
    for (unsigned t = tid; t < nb; t += 1024) hist[t] = 0;
    __syncthreads();

    for (int e = tid; e < A_TOTAL; e += 1024) {
      unsigned k = fkey(obj[base + e]);
      if (prefShift == 32 || (k >> prefShift) == prefix)
        atomicAdd(&hist[(k >> lo) & (nb - 1u)], 1u);
    }
    __syncthreads();

    // suffix-sum (Hillis-Steele): hist[t] := count of bucket >= t
    for (unsigned d = 1; d < nb; d <<= 1) {
      unsigned v0 = 0, v1 = 0;
      unsigned t0 = tid, t1 = tid + 1024;
      if (t0 < nb) v0 = hist[t0] + ((t0 + d < nb) ? hist[t0 + d] : 0u);
      if (t1 < nb) v1 = hist[t1] + ((t1 + d < nb) ? hist[t1 + d] : 0u);
      __syncthreads();
      if (t0 < nb) hist[t0] = v0;
      if (t1 < nb) hist[t1] = v1;
      __syncthreads();
    }

    // pick threshold bucket: hist[T] >= need > hist[T+1]
    for (unsigned t = tid; t < nb; t += 1024) {
      unsigned c  = hist[t];
      unsigned cn = (t + 1 < nb) ? hist[t + 1] : 0u;
      if (c >= need && cn < need) { s_sel = t; s_above = cn; }
    }
    __syncthreads();
    prefix = (prefix << bits_r[r]) | s_sel;
    need  -= s_above;
    prefShift = lo;
    __syncthreads();
  }
  const unsigned K = prefix;              // exact threshold key
  const unsigned slotbase = PRE_N - need; // #{key > K}

  // init pads + counters, then collect
  for (int t = tid; t < NPAD; t += 1024) { s_sc[t] = -INFINITY; s_idx[t] = 0x7fffffffu; }
  if (tid == 0) { s_cnt_gt = 0; s_cnt_eq = 0; }
  __syncthreads();
  for (int e = tid; e < A_TOTAL; e += 1024) {
    float f = obj[base + e];
    unsigned k = fkey(f);
    if (k > K) {
      unsigned p = atomicAdd(&s_cnt_gt, 1u);
      s_sc[p] = f; s_idx[p] = (unsigned)e;
    } else if (k == K) {
      unsigned q = atomicAdd(&s_cnt_eq, 1u);
      if (q < need) { s_sc[slotbase + q] = f; s_idx[slotbase + q] = (unsigned)e; }
    }
  }
  __syncthreads();

  // bitonic sort 1024 elements, descending key, tie -> ascending index
  for (unsigned k2 = 2; k2 <= NPAD; k2 <<= 1) {
    for (unsigned j = k2 >> 1; j > 0; j >>= 1) {
      unsigned ixj = tid ^ j;
      if (ixj > (unsigned)tid) {
        float   a  = s_sc[tid],  bb = s_sc[ixj];
        unsigned ia = s_idx[tid], ib = s_idx[ixj];
        unsigned ka = fkey(a), kb = fkey(bb);
        bool a_worse = (ka < kb) || (ka == kb && ia > ib);
        bool up = ((tid & k2) == 0);
        if (up ? a_worse : !a_worse) {
          s_sc[tid] = bb; s_sc[ixj] = a;
          s_idx[tid] = ib; s_idx[ixj] = ia;
        }
      }
      __syncthreads();
    }
  }
  for (int t = tid; t < NPAD; t += 1024) {
    tscore[(size_t)b * NPAD + t] = s_sc[t];
    tidx[(size_t)b * NPAD + t]   = s_idx[t];
  }
}

// =====================================================================
// Kernel 2: gather + decode + clamp + validity + level offset.
// Invalid / padded rows get area = BIG_AREA so they can never suppress.
// =====================================================================
__global__ __launch_bounds__(256) void decode_kernel(const float* __restrict__ deltas,
                                                     const float* __restrict__ anchors,
                                                     const unsigned* __restrict__ tidx,
                                                     float* __restrict__ nmsbox,
                                                     float* __restrict__ areaArr,
                                                     float* __restrict__ boxes,
                                                     unsigned* __restrict__ vflag) {
  const int gid = blockIdx.x * 256 + threadIdx.x;   // B*NPAD threads
  const int b = gid >> 10, i = gid & 1023;
  const size_t row = (size_t)b * NPAD + i;
  if (i >= PRE_N) {
    float4 z = {0.f, 0.f, 0.f, 0.f};
    ((float4*)nmsbox)[row] = z; ((float4*)boxes)[row] = z;
    areaArr[row] = BIG_AREA; vflag[row] = 0u;
    return;
  }
  const unsigned aidx = tidx[row];
  const float* dp = deltas + ((size_t)b * A_TOTAL + aidx) * 4;
  const float* ap = anchors + (size_t)aidx * 4;
  float dx = dp[0], dy = dp[1];
  float dw = fminf(dp[2], DWH_CLAMP), dh = fminf(dp[3], DWH_CLAMP);
  float w = ap[2] - ap[0], h = ap[3] - ap[1];
  float cx = ap[0] + 0.5f * w, cy = ap[1] + 0.5f * h;
  float pcx = cx + dx * w, pcy = cy + dy * h;
  float pw = expf(dw) * w, ph = expf(dh) * h;
  float x1 = pcx - 0.5f * pw, y1 = pcy - 0.5f * ph;
  float x2 = pcx + 0.5f * pw, y2 = pcy + 0.5f * ph;
  x1 = fminf(fmaxf(x1, 0.f), IMG_F); y1 = fminf(fmaxf(y1, 0.f), IMG_F);
  x2 = fminf(fmaxf(x2, 0.f), IMG_F); y2 = fminf(fmaxf(y2, 0.f), IMG_F);
  bool valid = ((x2 - x1) >= MIN_SZ) && ((y2 - y1) >= MIN_SZ);
  int lvl = (aidx >= 196608u) + (aidx >= 245760u) + (aidx >= 258048u) + (aidx >= 261120u);
  float off = (float)lvl * LVL_OFF;
  float4 nb = {x1 + off, y1 + off, x2 + off, y2 + off};
  float4 bb = {x1, y1, x2, y2};
  ((float4*)nmsbox)[row] = nb;
  ((float4*)boxes)[row]  = bb;
  areaArr[row] = valid ? (x2 - x1) * (y2 - y1) : BIG_AREA;
  vflag[row]   = valid ? 1u : 0u;
}

// =====================================================================
// Kernel 3: suppression bitmask. Block = (row-tile of 16, batch),
// 256 threads = 8 wave32. Boxes/areas async-staged into LDS; each wave
// handles 16x16 tiles: WMMA 16x16x4 F32 computes area[i]+area[j] for the
// whole tile (A row m = [a_i, 1, 0, 0], B = [[1..1],[a_j..],0,0]); VALU
// does min/max intersection; ballot -> 16-bit row masks -> ds_or in LDS.
// =====================================================================
__global__ __launch_bounds__(256) void iou_mask_kernel(const float* __restrict__ nmsbox,
                                                       const float* __restrict__ areaArr,
                                                       unsigned* __restrict__ mask) {
  __shared__ float    s_box[NPAD * 4];
  __shared__ float    s_area[NPAD];
  __shared__ unsigned s_row[16 * 32];

  const int tid = threadIdx.x;
  const int b   = blockIdx.y;
  const int i0  = blockIdx.x * 16;
  const int wave = tid >> 5, lane = tid & 31;
  const int n = lane & 15, half = lane >> 4;

  const float* gb = nmsbox  + (size_t)b * NPAD * 4;
  const float* ga = areaArr + (size_t)b * NPAD;
#if USE_ASYNC_LDS
  for (int r = tid; r < NPAD; r += 256)       ldsCopyB128(&s_box[r * 4], &gb[r * 4]);
  for (int r = tid; r < NPAD / 4; r += 256)   ldsCopyB128(&s_area[r * 4], &ga[r * 4]);
  ldsCopyWait();
#else
  for (int r = tid; r < NPAD; r += 256)       ((float4*)s_box)[r]  = ((const float4*)gb)[r];
  for (int r = tid; r < NPAD / 4; r += 256)   ((float4*)s_area)[r] = ((const float4*)ga)[r];
#endif
  for (int w = tid; w < 512; w += 256) s_row[w] = 0u;
  __syncthreads();

  // per-lane cached row boxes: this lane's rows are i0 + v + 8*half, v=0..7
  float rx1[8], ry1[8], rx2[8], ry2[8];
#if !USE_WMMA_F32
  float rar[8];
#endif
  for (int v = 0; v < 8; ++v) {
    int i = i0 + v + (half << 3);
    rx1[v] = s_box[i * 4 + 0]; ry1[v] = s_box[i * 4 + 1];
    rx2[v] = s_box[i * 4 + 2]; ry2[v] = s_box[i * 4 + 3];
#if !USE_WMMA_F32
    rar[v] = s_area[i];
#endif
  }
  // A operand: lanes 0-15 hold M=lane rows, K0/K1 = {area_i, 1}; lanes 16-31 K2/K3 = 0.
  v2f Av;
  Av.x = (half == 0) ? s_area[i0 + n] : 0.0f;
  Av.y = (half == 0) ? 1.0f : 0.0f;

  for (int t = 0; t < 8; ++t) {
    const int tj = wave + (t << 3);     // column tile 0..63
    const int j0 = tj << 4;
    const int j  = j0 + n;
    float jx1 = s_box[j * 4 + 0], jy1 = s_box[j * 4 + 1];
    float jx2 = s_box[j * 4 + 2], jy2 = s_box[j * 4 + 3];
    float ja  = s_area[j];

    v8f D = {};
#if USE_WMMA_F32
    // B operand (assumed K packing symmetric to documented A layout):
    // VGPR0: lanes0-15 = K0 row (ones), lanes16-31 = K2 (0)
    // VGPR1: lanes0-15 = K1 row (area_j), lanes16-31 = K3 (0)
    v2f Bv;
    Bv.x = (half == 0) ? 1.0f : 0.0f;
    Bv.y = (half == 0) ? ja   : 0.0f;
    D = __builtin_amdgcn_wmma_f32_16x16x4_f32(false, Av, false, Bv,
                                              (short)0, D, false, false);
#else
    for (int v = 0; v < 8; ++v) D[v] = rar[v] + ja;   // fallback: VALU area sum
#endif

    const int wword = j0 >> 5, wshift = (j0 & 16);
    for (int v = 0; v < 8; ++v) {
      int i = i0 + v + (half << 3);
      float xx1 = fmaxf(rx1[v], jx1), yy1 = fmaxf(ry1[v], jy1);
      float iw = fminf(rx2[v], jx2) - xx1;
      float ih = fminf(ry2[v], jy2) - yy1;
      float inter = iw * ih;
      bool pred = (iw > 0.f) && (ih > 0.f) && (inter > IOU_C * D[v]) && (j > i);
      unsigned m32 = (unsigned)__ballot((int)pred);   // wave32: low 32 bits
      if (lane == 0) atomicOr(&s_row[v * 32 + wword], (m32 & 0xFFFFu) << wshift);
      if (lane == 1) atomicOr(&s_row[(v + 8) * 32 + wword], (m32 >> 16) << wshift);
    }
  }
  __syncthreads();
  unsigned* gm = mask + ((size_t)(b << 10) + i0) * 32;
  for (int w = tid; w < 512; w += 256) gm[w] = s_row[w];
}

// =====================================================================
// Kernel 4: serial greedy NMS + output. One block per batch. The 128 KB
// suppression matrix is async-staged into LDS (320 KB/WGP on CDNA5);
// wave 0 holds the 1024-bit suppression state as 1 word/lane (wave32).
// =====================================================================
__global__ __launch_bounds__(256) void nms_out_kernel(const unsigned* __restrict__ mask,
                                                      const unsigned* __restrict__ vflag,
                                                      const float* __restrict__ boxes,
                                                      const float* __restrict__ tscore,
                                                      float* __restrict__ out) {
  extern __shared__ unsigned char k4smem[];
  unsigned* smask = (unsigned*)k4smem;                       // 1024*32 words = 128 KB
  unsigned* skeep = (unsigned*)(k4smem + (size_t)NPAD * 32 * 4);

  const int tid = threadIdx.x;
  const int b   = blockIdx.x;
  const unsigned* gm = mask + (size_t)b * NPAD * 32;
#if USE_ASYNC_LDS
  for (int c = tid; c < NPAD * 32 / 4; c += 256) ldsCopyB128(&smask[c * 4], &gm[c * 4]);
  ldsCopyWait();
#else
  for (int c = tid; c < NPAD * 32 / 4; c += 256) ((uint4*)smask)[c] = ((const uint4*)gm)[c];
#endif
  __syncthreads();

  if (tid < 32) {
    const int lane = tid;
    unsigned vw = 0;
    const unsigned* vf = vflag + (size_t)b * NPAD + lane * 32;
    for (int t = 0; t < 32; ++t) vw |= (vf[t] & 1u) << t;
    unsigned supp = ~vw;                        // invalid rows start suppressed
    for (int i = 0; i < PRE_N; ++i) {
      unsigned w = __shfl(supp, i >> 5);        // broadcast owner lane's word
      if (!((w >> (i & 31)) & 1u))
        supp |= smask[i * 32 + lane];           // apply row i (j>i, valid_j folded in)
    }
    skeep[lane] = ~supp;
  }
  __syncthreads();

  for (int i = tid; i < PRE_N; i += 256) {
    bool kp = (skeep[i >> 5] >> (i & 31)) & 1u;
    const float* bx = boxes + ((size_t)b * NPAD + i) * 4;
    float sc = tscore[(size_t)b * NPAD + i];
    float* o = out + ((size_t)b * PRE_N + i) * 5;
    o[0] = kp ? bx[0] : 0.f;
    o[1] = kp ? bx[1] : 0.f;
    o[2] = kp ? bx[2] : 0.f;
    o[3] = kp ? bx[3] : 0.f;
    o[4] = kp ? sc    : 0.f;
  }
}

// =====================================================================
extern "C" void kernel_launch(void* const* d_in, const int* in_sizes, int n_in,
                              void* d_out, int out_size, void* d_ws, size_t ws_size,
                              hipStream_t stream) {
  (void)in_sizes; (void)n_in; (void)out_size; (void)ws_size;
  const float* obj     = (const float*)d_in[0];   // [B, A_TOTAL]
  const float* deltas  = (const float*)d_in[1];   // [B, A_TOTAL, 4]
  const float* anchors = (const float*)d_in[2];   // [A_TOTAL, 4]
  float* out = (float*)d_out;                     // [B, 1000, 5]

  char* ws = (char*)d_ws;
  size_t off = 0;
  auto carve = [&](size_t bytes) -> void* {
    void* p = ws + off;
    off = (off + bytes + 255) & ~(size_t)255;
    return p;
  };
  unsigned* tidx    = (unsigned*)carve((size_t)BATCH * NPAD * 4);
  float*    tscore  = (float*)   carve((size_t)BATCH * NPAD * 4);
  float*    nmsbox  = (float*)   carve((size_t)BATCH * NPAD * 16);
  float*    areaArr = (float*)   carve((size_t)BATCH * NPAD * 4);
  float*    boxes   = (float*)   carve((size_t)BATCH * NPAD * 16);
  unsigned* vflag   = (unsigned*)carve((size_t)BATCH * NPAD * 4);
  unsigned* maskBuf = (unsigned*)carve((size_t)BATCH * NPAD * 32 * 4);

  topk_kernel<<<dim3(BATCH), dim3(1024), 0, stream>>>(obj, tscore, tidx);
  decode_kernel<<<dim3(BATCH * NPAD / 256), dim3(256), 0, stream>>>(
      deltas, anchors, tidx, nmsbox, areaArr, boxes, vflag);
  iou_mask_kernel<<<dim3(NPAD / 16, BATCH), dim3(256), 0, stream>>>(
      nmsbox, areaArr, maskBuf);
  nms_out_kernel<<<dim3(BATCH), dim3(256), (size_t)NPAD * 32 * 4 + 256, stream>>>(
      maskBuf, vflag, boxes, tscore, out);
}